// ImprovedYoloLoss_83854941487214
// MI455X (gfx1250) — compile-verified
//
#include <hip/hip_runtime.h>

typedef __attribute__((ext_vector_type(2))) float v2f;
typedef __attribute__((ext_vector_type(8))) float v8f;

#define NC    80
#define CCH   144   // 64 box/dfl channels + 80 class channels
#define NBATCH 32
#define NT     64   // targets per image

static constexpr float INV_TOTAL = 1.0f / (3.0f * NBATCH * NT);  // 1/6144

// stable softplus(x) = max(x,0) + log1p(exp(-|x|))
__device__ __forceinline__ float softplusf(float x) {
  return fmaxf(x, 0.0f) + log1pf(__expf(-fabsf(x)));
}

__device__ __forceinline__ float smooth_l1(float d) {
  float ad = fabsf(d);
  return (ad < 1.0f) ? (0.5f * d * d) : (ad - 0.5f);
}

__device__ __forceinline__ void block_reduce_store(float v, float* __restrict__ dst) {
  __shared__ float sm[256];
  int tid = threadIdx.x;
  sm[tid] = v;
  __syncthreads();
#pragma unroll
  for (int ofs = 128; ofs >= 1; ofs >>= 1) {
    if (tid < ofs) sm[tid] += sm[tid + ofs];
    __syncthreads();
  }
  if (tid == 0) dst[blockIdx.x] = sm[0];
}

// ---------------------------------------------------------------------------
// Kernel 1: negative-objectness term over every grid cell of all 3 scales.
// contribution = 0.1 * (neg > 0.9) * softplus(pred[b,0,y,x]) * LAMBDA_OBJ / 6144
// 268,800 cells total = exactly 1050 blocks * 256 threads.
// ---------------------------------------------------------------------------
__global__ void __launch_bounds__(256)
yolo_neg_kernel(const float* __restrict__ p0, const float* __restrict__ n0,
                const float* __restrict__ p1, const float* __restrict__ n1,
                const float* __restrict__ p2, const float* __restrict__ n2,
                float* __restrict__ partial) {
  const int cum0 = NBATCH * 400;          // 12800
  const int cum1 = cum0 + NBATCH * 1600;  // 64000
  const int cum2 = cum1 + NBATCH * 6400;  // 268800

  int t = blockIdx.x * 256 + threadIdx.x;
  float acc = 0.0f;
  if (t < cum2) {
    const float* pp;
    const float* nn;
    int hw_sz, local;
    if (t < cum0)      { pp = p0; nn = n0; hw_sz = 400;  local = t; }
    else if (t < cum1) { pp = p1; nn = n1; hw_sz = 1600; local = t - cum0; }
    else               { pp = p2; nn = n2; hw_sz = 6400; local = t - cum1; }
    int b  = local / hw_sz;
    int hw = local - b * hw_sz;
    float r = nn[b * hw_sz + hw];
    if (r > 0.9f) {
      float x = pp[(size_t)b * CCH * hw_sz + hw];   // channel 0 plane, contiguous
      acc = 0.1f * softplusf(x);
    }
  }
  block_reduce_store(acc * INV_TOTAL, partial);
}

// ---------------------------------------------------------------------------
// Kernel 2: per-target box / cls / positive-obj terms. 6144 targets = 24 blocks.
// ---------------------------------------------------------------------------
__global__ void __launch_bounds__(256)
yolo_tgt_kernel(const float* __restrict__ p0, const float* __restrict__ p1,
                const float* __restrict__ p2,
                const int* __restrict__ tgt_cls, const float* __restrict__ tgt_boxes,
                float* __restrict__ partial) {
  int t = blockIdx.x * 256 + threadIdx.x;        // 0..6143
  int s = t / (NBATCH * NT);                     // scale
  int r = t - s * (NBATCH * NT);                 // b*NT + n
  int b = r / NT;

  const float* pp = (s == 0) ? p0 : ((s == 1) ? p1 : p2);
  int W = (s == 0) ? 20 : ((s == 1) ? 40 : 80);
  int hw_sz = W * W;

  const float* box = tgt_boxes + (size_t)r * 4;
  float gcx = box[0] * (float)W;
  float gcy = box[1] * (float)W;
  float gw  = box[2] * (float)W;
  float gh  = box[3] * (float)W;
  int gx = min(max((int)gcx, 0), W - 1);         // trunc == astype(int32) for x>=0
  int gy = min(max((int)gcy, 0), W - 1);
  int flat = gy * W + gx;

  size_t base = (size_t)b * CCH * hw_sz + flat;

  float pbox[4];
#pragma unroll
  for (int c = 0; c < 4; ++c) pbox[c] = pp[base + (size_t)c * hw_sz];

  float tx = gcx - (float)gx;
  float ty = gcy - (float)gy;
  float tw = logf(gw + 1e-16f);
  float th = logf(gh + 1e-16f);
  float lbox = 0.25f * (smooth_l1(pbox[0] - tx) + smooth_l1(pbox[1] - ty) +
                        smooth_l1(pbox[2] - tw) + smooth_l1(pbox[3] - th));

  int cls = tgt_cls[r];
  const float* pc = pp + base + (size_t)64 * hw_sz;
  float lcls = 0.0f;
  for (int j = 0; j < NC; ++j) {
    float x = pc[(size_t)j * hw_sz];
    lcls += softplusf(x) - ((j == cls) ? x : 0.0f);
  }

  float lobj = softplusf(-pbox[0]);

  float contrib = (7.5f * lbox + lobj + 0.5f * lcls) * INV_TOTAL;
  block_reduce_store(contrib, partial);
}

// ---------------------------------------------------------------------------
// Kernel 3: single wave32 final reduction. Strided lane sums, then a
// V_WMMA_F32_16X16X4_F32 with A = ones: every row of D equals the column-sum
// vector of B, so summing c[0] over all 32 lanes yields 2x the grand total.
// EXEC is all 1s for the full wave as the WMMA restriction requires.
// ---------------------------------------------------------------------------
__global__ void __launch_bounds__(32)
yolo_final_kernel(const float* __restrict__ partial, int npart, float* __restrict__ out) {
  int lane = threadIdx.x;
  float s = 0.0f;
  for (int i = lane; i < npart; i += 32) s += partial[i];

  v2f a; a[0] = 1.0f; a[1] = 1.0f;        // A (16x4) all ones
  v2f bm; bm[0] = s;  bm[1] = 0.0f;       // each lane sum in a unique B slot
  v8f c = {};
  c = __builtin_amdgcn_wmma_f32_16x16x4_f32(
      /*neg_a=*/false, a, /*neg_b=*/false, bm,
      /*c_mod=*/(short)0, c, /*reuse_a=*/false, /*reuse_b=*/false);

  float v = c[0];                          // rows M=0 (lanes 0-15) and M=8 (16-31)
#pragma unroll
  for (int m = 16; m >= 1; m >>= 1) v += __shfl_xor(v, m, 32);
  if (lane == 0) out[0] = 0.5f * v;        // both rows identical -> halve
}

// ---------------------------------------------------------------------------
extern "C" void kernel_launch(void* const* d_in, const int* in_sizes, int n_in,
                              void* d_out, int out_size, void* d_ws, size_t ws_size,
                              hipStream_t stream) {
  const float* p0   = (const float*)d_in[0];
  const float* n0   = (const float*)d_in[1];
  const float* p1   = (const float*)d_in[2];
  const float* n1   = (const float*)d_in[3];
  const float* p2   = (const float*)d_in[4];
  const float* n2   = (const float*)d_in[5];
  const int*   tcls = (const int*)d_in[6];
  const float* tbox = (const float*)d_in[7];

  float* ws = (float*)d_ws;
  const int NEG_BLOCKS = 1050;   // 268800 / 256
  const int TGT_BLOCKS = 24;     // 6144 / 256
  const int NPART = NEG_BLOCKS + TGT_BLOCKS;  // 1074 floats of scratch

  yolo_neg_kernel<<<NEG_BLOCKS, 256, 0, stream>>>(p0, n0, p1, n1, p2, n2, ws);
  yolo_tgt_kernel<<<TGT_BLOCKS, 256, 0, stream>>>(p0, p1, p2, tcls, tbox, ws + NEG_BLOCKS);
  yolo_final_kernel<<<1, 32, 0, stream>>>(ws, NPART, (float*)d_out);
}